// GATLayer_188978561163
// MI455X (gfx1250) — compile-verified
//
#include <hip/hip_runtime.h>
#include <hip/hip_bf16.h>
#include <math.h>

typedef __attribute__((ext_vector_type(2))) float v2f;
typedef __attribute__((ext_vector_type(8))) float v8f;

#define IN_FEATS 128
#define HEADS    4
#define FTOT     64      // HEADS * OUT_FEATS
#define ALPHA    0.2f

// ---- monotone float<->uint encoding for atomicMax on floats ----
__device__ __forceinline__ unsigned encf(float f) {
    unsigned u = __float_as_uint(f);
    return (u & 0x80000000u) ? ~u : (u | 0x80000000u);
}
__device__ __forceinline__ float decf(unsigned u) {
    u = (u & 0x80000000u) ? (u & 0x7FFFFFFFu) : ~u;
    return __uint_as_float(u);
}

// ------------------------------------------------------------------
// Kernel 1: Wh = h @ W   (50000x128 * 128x64), exact fp32 via WMMA.
// Block = 128 threads (4 waves). Each block: 16 rows x 64 cols.
// Wave w computes the 16x16 tile at cols [16w, 16w+16).
// A tile (16x128) and all of W (128x64) staged in LDS (40 KB).
// ------------------------------------------------------------------
__global__ void __launch_bounds__(128)
k_gemm_wmma(const float* __restrict__ h, const float* __restrict__ W,
            float* __restrict__ Wh, int N) {
    __shared__ float sW[IN_FEATS * FTOT];   // 32 KB
    __shared__ float sA[16 * IN_FEATS];     // 8 KB

    const int tid   = threadIdx.x;
    const int block = blockIdx.x;
    const int row0  = block * 16;

    // stage W: 8192 floats = 2048 float4
    {
        const float4* W4  = (const float4*)W;
        float4*       sW4 = (float4*)sW;
        for (int i = tid; i < (IN_FEATS * FTOT) / 4; i += 128) sW4[i] = W4[i];
    }
    // stage A tile: 16*128 floats = 512 float4 (32 float4 per row)
    {
        const float4* h4  = (const float4*)h;
        float4*       sA4 = (float4*)sA;
        for (int i = tid; i < 512; i += 128) {
            int r  = i >> 5;                 // row in tile
            int c4 = i & 31;                 // float4 within row
            int gr = row0 + r; if (gr > N - 1) gr = N - 1;
            sA4[i] = h4[(size_t)gr * (IN_FEATS / 4) + c4];
        }
    }
    __syncthreads();

    const int wave  = tid >> 5;          // 0..3 -> col tile
    const int lane  = tid & 31;
    const int r     = lane & 15;         // row within tile / N index for B
    const int khalf = lane >> 4;         // 0 or 1 (K-half split at lane 16)
    const int col0  = wave * 16;

    v8f c = {};
    #pragma unroll
    for (int k0 = 0; k0 < IN_FEATS; k0 += 4) {
        // A 16x4 frag (ISA 7.12.2): VGPR0 = K {k0 | k0+2}, VGPR1 = K {k0+1 | k0+3}
        v2f a;
        {
            float2 t = *(const float2*)&sA[r * IN_FEATS + k0 + 2 * khalf];
            a.x = t.x; a.y = t.y;
        }
        // B 4x16 frag: rows striped across lanes; VGPR j holds K = k0 + 2j + khalf
        v2f b;
        b.x = sW[(k0 + khalf    ) * FTOT + col0 + r];
        b.y = sW[(k0 + khalf + 2) * FTOT + col0 + r];
        c = __builtin_amdgcn_wmma_f32_16x16x4_f32(false, a, false, b,
                                                  (short)0, c, false, false);
    }

    // C/D layout: VGPR j -> row j (lanes 0-15) / row j+8 (lanes 16-31), N = lane&15
    const int rbase = row0 + (khalf ? 8 : 0);
    const int cidx  = col0 + r;
    #pragma unroll
    for (int j = 0; j < 8; ++j) {
        int gr = rbase + j;
        if (gr < N) Wh[(size_t)gr * FTOT + cidx] = c[j];
    }
}

// ------------------------------------------------------------------
// Kernel 0: zero output + tiny accumulators
// ------------------------------------------------------------------
__global__ void k_init(float* __restrict__ out, unsigned* __restrict__ maxenc,
                       float* __restrict__ sums, int n_out) {
    int i = blockIdx.x * blockDim.x + threadIdx.x;
    if (i < n_out) out[i] = 0.0f;
    if (i < HEADS) { maxenc[i] = 0u; sums[i] = 0.0f; }
}

// ------------------------------------------------------------------
// Kernel 2: per-node attention scalars s_src/s_dst  [N, HEADS]
// ------------------------------------------------------------------
__global__ void k_scores(const float* __restrict__ Wh,
                         const float* __restrict__ a_src,
                         const float* __restrict__ a_dst,
                         float* __restrict__ s_src, float* __restrict__ s_dst,
                         int N) {
    int n = blockIdx.x * blockDim.x + threadIdx.x;
    if (n >= N) return;
    const float4* row = (const float4*)(Wh + (size_t)n * FTOT);
    const float4* as4 = (const float4*)a_src;
    const float4* ad4 = (const float4*)a_dst;
    #pragma unroll
    for (int hh = 0; hh < HEADS; ++hh) {
        float ss = 0.f, sd = 0.f;
        #pragma unroll
        for (int q = 0; q < 4; ++q) {
            float4 v = row[hh * 4 + q];
            float4 a = as4[hh * 4 + q];
            float4 b = ad4[hh * 4 + q];
            ss += v.x * a.x + v.y * a.y + v.z * a.z + v.w * a.w;
            sd += v.x * b.x + v.y * b.y + v.z * b.z + v.w * b.w;
        }
        s_src[(size_t)n * HEADS + hh] = ss;
        s_dst[(size_t)n * HEADS + hh] = sd;
    }
}

// ------------------------------------------------------------------
// Kernel 3: per-head global max of leakyrelu(e) over all edges
// ------------------------------------------------------------------
__global__ void k_edge_max(const long long* __restrict__ ei,
                           const float* __restrict__ s_src,
                           const float* __restrict__ s_dst,
                           unsigned* __restrict__ gmax, long long E) {
    __shared__ unsigned sm[HEADS];
    const int t = threadIdx.x;
    const long long e = (long long)blockIdx.x * blockDim.x + t;
    if (t < HEADS) sm[t] = 0u;
    __syncthreads();
    if (e < E) {
        int src = (int)ei[e];
        int dst = (int)ei[E + e];
        #pragma unroll
        for (int hh = 0; hh < HEADS; ++hh) {
            float v = s_src[(size_t)src * HEADS + hh] + s_dst[(size_t)dst * HEADS + hh];
            v = v > 0.f ? v : ALPHA * v;
            atomicMax(&sm[hh], encf(v));
        }
    }
    __syncthreads();
    if (t < HEADS) atomicMax(&gmax[t], sm[t]);
}

// ------------------------------------------------------------------
// Kernel 4: exp(e - max), store unnormalized attention, per-head sums
// ------------------------------------------------------------------
__global__ void k_edge_sum(const long long* __restrict__ ei,
                           const float* __restrict__ s_src,
                           const float* __restrict__ s_dst,
                           const unsigned* __restrict__ gmax,
                           float* __restrict__ sums,
                           float* __restrict__ attbuf, long long E) {
    __shared__ float ssum[HEADS];
    const int t = threadIdx.x;
    const long long e = (long long)blockIdx.x * blockDim.x + t;
    if (t < HEADS) ssum[t] = 0.0f;
    __syncthreads();
    if (e < E) {
        int src = (int)ei[e];
        int dst = (int)ei[E + e];
        #pragma unroll
        for (int hh = 0; hh < HEADS; ++hh) {
            float v = s_src[(size_t)src * HEADS + hh] + s_dst[(size_t)dst * HEADS + hh];
            v = v > 0.f ? v : ALPHA * v;
            float ex = __expf(v - decf(gmax[hh]));
            attbuf[(size_t)e * HEADS + hh] = ex;
            atomicAdd(&ssum[hh], ex);
        }
    }
    __syncthreads();
    if (t < HEADS) atomicAdd(&sums[t], ssum[t]);
}

__global__ void k_finalize(const float* __restrict__ sums, float* __restrict__ inv) {
    int t = threadIdx.x;
    if (t < HEADS) inv[t] = 1.0f / sums[t];
}

// ------------------------------------------------------------------
// Kernel 5: scatter  out[dst, f] += att[e, f/16] * Wh[src, f]
// one thread per (edge, feature)
// ------------------------------------------------------------------
__global__ void k_scatter(const long long* __restrict__ ei,
                          const float* __restrict__ Wh,
                          const float* __restrict__ attbuf,
                          const float* __restrict__ inv,
                          float* __restrict__ out, long long E) {
    long long t = (long long)blockIdx.x * blockDim.x + threadIdx.x;
    long long e = t >> 6;
    if (e >= E) return;
    int f  = (int)(t & 63);
    int hh = f >> 4;
    int src = (int)ei[e];
    int dst = (int)ei[E + e];
    float a = attbuf[(size_t)e * HEADS + hh] * inv[hh];
    float v = a * Wh[(size_t)src * FTOT + f];
    unsafeAtomicAdd(&out[(size_t)dst * FTOT + f], v);
}

__global__ void k_relu(float* __restrict__ out, int n) {
    int i = blockIdx.x * blockDim.x + threadIdx.x;
    if (i < n) { float v = out[i]; out[i] = v > 0.f ? v : 0.f; }
}

// ------------------------------------------------------------------
extern "C" void kernel_launch(void* const* d_in, const int* in_sizes, int n_in,
                              void* d_out, int out_size, void* d_ws, size_t ws_size,
                              hipStream_t stream) {
    const long long* ei    = (const long long*)d_in[0];   // int64 [2, E]
    const float*     h     = (const float*)d_in[1];       // [N, 128]
    const float*     W     = (const float*)d_in[2];       // [128, 64]
    const float*     a_src = (const float*)d_in[3];       // [64]
    const float*     a_dst = (const float*)d_in[4];       // [64]
    float*           out   = (float*)d_out;               // [N, 64]

    const long long E = in_sizes[0] / 2;
    const int       N = in_sizes[1] / IN_FEATS;

    // ---- workspace carve ----
    float*    Wh     = (float*)d_ws;                       // N*64
    float*    s_src  = Wh + (size_t)N * FTOT;              // N*4
    float*    s_dst  = s_src + (size_t)N * HEADS;          // N*4
    float*    attbuf = s_dst + (size_t)N * HEADS;          // E*4
    unsigned* maxenc = (unsigned*)(attbuf + (size_t)E * HEADS); // 4
    float*    sums   = (float*)(maxenc + HEADS);           // 4
    float*    inv    = sums + HEADS;                       // 4

    const int n_out = N * FTOT;

    k_init<<<(n_out + 255) / 256, 256, 0, stream>>>(out, maxenc, sums, n_out);

    k_gemm_wmma<<<(N + 15) / 16, 128, 0, stream>>>(h, W, Wh, N);

    k_scores<<<(N + 255) / 256, 256, 0, stream>>>(Wh, a_src, a_dst, s_src, s_dst, N);

    const int eb = (int)((E + 255) / 256);
    k_edge_max<<<eb, 256, 0, stream>>>(ei, s_src, s_dst, maxenc, E);
    k_edge_sum<<<eb, 256, 0, stream>>>(ei, s_src, s_dst, maxenc, sums, attbuf, E);
    k_finalize<<<1, 32, 0, stream>>>(sums, inv);

    const long long tot = E * 64;
    k_scatter<<<(int)((tot + 255) / 256), 256, 0, stream>>>(ei, Wh, attbuf, inv, out, E);

    k_relu<<<(n_out + 255) / 256, 256, 0, stream>>>(out, n_out);
}